// QuantumQuanvolutionFilter_65481071406708
// MI455X (gfx1250) — compile-verified
//
#include <hip/hip_runtime.h>
#include <hip/hip_bf16.h>

// Closed form of the 4-qubit quanvolution circuit:
//   m_i = cos(params[i]) * cos(theta_i), theta = 2x2 patch pixels (row-major)
//   z0 = m1*m2*m3, z1 = m0*m1, z2 = m0*m1*m2, z3 = m0*m1*m2*m3
// Memory-bound: ~25.7 MB total traffic -> ~1.1 us at 23.3 TB/s HBM.
// All global traffic is 16B-aligned b128, fully coalesced. No matrix
// structure exists in this op, so WMMA is intentionally not used.
//
// Addressing: thread idx = b*98 + r*7 + c2 (c2 = patch-pair column).
// Since 98 % 7 == 0, c2 = idx % 7; input float offset = idx*8 - 4*c2,
// output float offset = idx*8. No division needed.
//
// IL images are processed per thread at a stride of (B/IL)*784 floats,
// amortizing the per-thread preamble/index cost (issue rate is near
// co-limiting with HBM bandwidth for this tiny per-thread body) with
// zero additional address arithmetic.

typedef float v4f __attribute__((ext_vector_type(4)));

__device__ __forceinline__ void quanv_pair(
    const v4f row0, const v4f row1,
    float cp0, float cp1, float cp2, float cp3,
    v4f& oA, v4f& oB)
{
    {
        const float m0 = cp0 * __cosf(row0.x);
        const float m1 = cp1 * __cosf(row0.y);
        const float m2 = cp2 * __cosf(row1.x);
        const float m3 = cp3 * __cosf(row1.y);
        const float m01 = m0 * m1;
        const float m23 = m2 * m3;
        oA.x = m1 * m23;      // <Z0> = m1*m2*m3
        oA.y = m01;           // <Z1> = m0*m1
        oA.z = m01 * m2;      // <Z2> = m0*m1*m2
        oA.w = m01 * m23;     // <Z3> = m0*m1*m2*m3
    }
    {
        const float m0 = cp0 * __cosf(row0.z);
        const float m1 = cp1 * __cosf(row0.w);
        const float m2 = cp2 * __cosf(row1.z);
        const float m3 = cp3 * __cosf(row1.w);
        const float m01 = m0 * m1;
        const float m23 = m2 * m3;
        oB.x = m1 * m23;
        oB.y = m01;
        oB.z = m01 * m2;
        oB.w = m01 * m23;
    }
}

template <int IL>
__global__ __launch_bounds__(256) void quanv_kernel(
    const float* __restrict__ x,       // [B, 784] = [B,1,28,28] flattened
    const float* __restrict__ params,  // [4]
    float* __restrict__ out,           // [B, 784]
    unsigned int nthreads,             // (B/IL) * 98
    unsigned int imgStride)            // (B/IL) * 784 floats
{
    const unsigned int idx = blockIdx.x * blockDim.x + threadIdx.x;
    if (idx >= nthreads) return;

    // cos(params): uniform scalar load, 4 trans ops (free; memory-bound).
    const v4f pv = *reinterpret_cast<const v4f*>(params);
    const float cp0 = __cosf(pv.x);
    const float cp1 = __cosf(pv.y);
    const float cp2 = __cosf(pv.z);
    const float cp3 = __cosf(pv.w);

    // c2 = patch-pair column (0..6); single unsigned mod, no other idx math.
    const unsigned int c2 = idx % 7u;

    const size_t inOff  = (size_t)idx * 8 - c2 * 4;  // 16B-aligned b128
    const size_t outOff = (size_t)idx * 8;           // 16B-aligned b128

    // Issue all loads first (one clause, ASYNC latency overlap), then math.
    v4f r0[IL], r1[IL];
#pragma unroll
    for (int i = 0; i < IL; ++i) {
        const float* img = x + inOff + (size_t)i * imgStride;
        r0[i] = *reinterpret_cast<const v4f*>(img);       // y=2r,   x=4c2..4c2+3
        r1[i] = *reinterpret_cast<const v4f*>(img + 28);  // y=2r+1, x=4c2..4c2+3
    }

#pragma unroll
    for (int i = 0; i < IL; ++i) {
        v4f oA, oB;
        quanv_pair(r0[i], r1[i], cp0, cp1, cp2, cp3, oA, oB);
        v4f* o = reinterpret_cast<v4f*>(out + outOff + (size_t)i * imgStride);
        o[0] = oA;   // patches (r, 2*c2) and (r, 2*c2+1): 8 contiguous floats
        o[1] = oB;
    }
}

extern "C" void kernel_launch(void* const* d_in, const int* in_sizes, int n_in,
                              void* d_out, int out_size, void* d_ws, size_t ws_size,
                              hipStream_t stream) {
    const float* x      = (const float*)d_in[0];   // [B,1,28,28] float32
    const float* params = (const float*)d_in[1];   // [4] float32
    float* out = (float*)d_out;                    // [B,784] float32

    const int B = in_sizes[0] / 784;               // 4096
    const int block = 256;                         // 8 wave32 per workgroup

    if ((B & 1) == 0) {
        const unsigned int nthreads  = (unsigned int)(B / 2) * 98u;   // 200,704
        const unsigned int imgStride = (unsigned int)(B / 2) * 784u;
        const int grid = (int)((nthreads + block - 1) / block);
        quanv_kernel<2><<<grid, block, 0, stream>>>(x, params, out, nthreads, imgStride);
    } else {
        const unsigned int nthreads = (unsigned int)B * 98u;
        const int grid = (int)((nthreads + block - 1) / block);
        quanv_kernel<1><<<grid, block, 0, stream>>>(x, params, out, nthreads, 0u);
    }
}